// AnchorSelector_50233937494428
// MI455X (gfx1250) — compile-verified
//
#include <hip/hip_runtime.h>
#include <hip/hip_bf16.h>

typedef float v2f __attribute__((ext_vector_type(2)));
typedef float v8f __attribute__((ext_vector_type(8)));

#define B_DIM    8
#define N_LOC    21824
#define C_DIM    256
#define A_DIM    9
#define K_SEL    1000
#define M_FLAT   (N_LOC * A_DIM)        /* 196416 per-batch candidates */
#define BN_TOT   (B_DIM * N_LOC)        /* 174592 locations */
#define NTILES   (BN_TOT / 16)          /* 10912, exact */
#define TOTAL_SEL (B_DIM * K_SEL)       /* 8000 */

/* workspace layout (bytes) */
#define KEYS_OFF  ((size_t)0)
#define KTH_OFF   ((size_t)B_DIM * M_FLAT * 4)     /* 6,285,312 */
#define ACNT_OFF  (KTH_OFF + 64)
#define SRC_OFF   (ACNT_OFF + 128)
#define ASLOT_OFF (SRC_OFF + (size_t)TOTAL_SEL * 4)

/* d_out layout (floats) */
#define OUT_B_OFF 2048000
#define OUT_F_OFF 2056000
#define OUT_A_OFF 2064000

__device__ __forceinline__ unsigned order_key(float f) {
    unsigned u = __float_as_uint(f);
    return (u & 0x80000000u) ? ~u : (u | 0x80000000u);
}

/* ---- kernel 0: zero per-anchor counters ---- */
__global__ void k_init(int* __restrict__ acnt) {
    if (threadIdx.x < 16) acnt[threadIdx.x] = 0;
}

/* ---- kernel 1: logits via fp32 WMMA, emit order-mapped keys ----
   One wave per 16-location tile; B operand = w_logits zero-padded to 16 cols in LDS. */
__global__ __launch_bounds__(256) void k_logits(const float* __restrict__ feats,
                                                const float* __restrict__ w_logits,
                                                const float* __restrict__ b_logits,
                                                unsigned* __restrict__ keys) {
    __shared__ float wl[C_DIM * 16];
    for (int i = threadIdx.x; i < C_DIM * 16; i += 256) {
        int c = i >> 4, a = i & 15;
        wl[i] = (a < A_DIM) ? w_logits[c * A_DIM + a] : 0.0f;
    }
    __syncthreads();

    int wave  = threadIdx.x >> 5;
    int lane  = threadIdx.x & 31;
    int tile  = blockIdx.x * 8 + wave;
    if (tile >= NTILES) return;
    int bn0   = tile * 16;
    int mrow  = lane & 15;
    int khalf = lane >> 4;
    const float* frow = feats + (size_t)(bn0 + mrow) * C_DIM + 2 * khalf;

    v8f acc = {0.f, 0.f, 0.f, 0.f, 0.f, 0.f, 0.f, 0.f};
    for (int k0 = 0; k0 < C_DIM; k0 += 4) {
        v2f av = *(const v2f*)(frow + k0);
        int kb = k0 + 2 * khalf;
        v2f bv;
        bv.x = wl[kb * 16 + mrow];
        bv.y = wl[(kb + 1) * 16 + mrow];
        acc = __builtin_amdgcn_wmma_f32_16x16x4_f32(false, av, false, bv,
                                                    (short)0, acc, false, false);
    }
    if (mrow < A_DIM) {
        float bias = b_logits[mrow];
        for (int v = 0; v < 8; v++) {
            int row = bn0 + v + 8 * khalf;
            keys[(size_t)row * A_DIM + mrow] = order_key(acc[v] + bias);
        }
    }
}

/* ---- kernel 2: per-batch radix-select of k-th largest key ---- */
__global__ __launch_bounds__(1024) void k_select(const unsigned* __restrict__ keys,
                                                 unsigned* __restrict__ kth) {
    __shared__ int hist[256];
    __shared__ unsigned s_prefix;
    __shared__ int s_rem;
    int b = blockIdx.x;
    if (threadIdx.x == 0) { s_prefix = 0u; s_rem = K_SEL; }
    __syncthreads();
    for (int pass = 0; pass < 4; pass++) {
        for (int i = threadIdx.x; i < 256; i += 1024) hist[i] = 0;
        __syncthreads();
        int shift = 24 - 8 * pass;
        unsigned prefix = s_prefix;
        for (int i = threadIdx.x; i < M_FLAT; i += 1024) {
            unsigned key = keys[(size_t)b * M_FLAT + i];
            bool match = (pass == 0) || ((key >> (shift + 8)) == prefix);
            if (match) atomicAdd(&hist[(key >> shift) & 0xFF], 1);
        }
        __syncthreads();
        if (threadIdx.x == 0) {
            int cum = 0, chosen = 0;
            for (int v = 255; v >= 0; v--) {
                if (cum + hist[v] >= s_rem) { chosen = v; break; }
                cum += hist[v];
            }
            s_rem -= cum;
            s_prefix = (s_prefix << 8) | (unsigned)chosen;
        }
        __syncthreads();
    }
    if (threadIdx.x == 0) kth[b] = s_prefix;
}

/* ---- kernel 3: compact winners, tie-break like lax.top_k, rank-sort, emit ids ---- */
__global__ __launch_bounds__(1024) void k_compact(const unsigned* __restrict__ keys,
                                                  const unsigned* __restrict__ kth,
                                                  int* __restrict__ acnt,
                                                  int* __restrict__ aslot,
                                                  int* __restrict__ slot_src,
                                                  float* __restrict__ out_b,
                                                  float* __restrict__ out_f,
                                                  float* __restrict__ out_a) {
    __shared__ unsigned s_key[K_SEL];
    __shared__ int s_idx[K_SEL];
    __shared__ int e_idx[2048];
    __shared__ int cnt_gt, cnt_eq;
    int b = blockIdx.x;
    unsigned kv = kth[b];
    if (threadIdx.x == 0) { cnt_gt = 0; cnt_eq = 0; }
    __syncthreads();
    for (int i = threadIdx.x; i < M_FLAT; i += 1024) {
        unsigned key = keys[(size_t)b * M_FLAT + i];
        if (key > kv) {
            int p = atomicAdd(&cnt_gt, 1);
            if (p < K_SEL) { s_key[p] = key; s_idx[p] = i; }
        } else if (key == kv) {
            int p = atomicAdd(&cnt_eq, 1);
            if (p < 2048) e_idx[p] = i;
        }
    }
    __syncthreads();
    int G = min(cnt_gt, K_SEL);
    int E = min(cnt_eq, 2048);
    int need = K_SEL - G;
    /* ties: take lowest indices first (top_k tie rule) */
    for (int e = threadIdx.x; e < E; e += 1024) {
        int my = e_idx[e], r = 0;
        for (int j = 0; j < E; j++) r += (e_idx[j] < my);
        if (r < need) { s_key[G + r] = kv; s_idx[G + r] = my; }
    }
    __syncthreads();
    int t = threadIdx.x;
    if (t < K_SEL) {
        unsigned mk = s_key[t];
        int mi = s_idx[t];
        int rank = 0;
        for (int j = 0; j < K_SEL; j++) {
            unsigned ok = s_key[j]; int oi = s_idx[j];
            rank += (ok > mk) || (ok == mk && oi < mi);
        }
        int id = mi;
        int a  = id % A_DIM;
        int n  = id / A_DIM;
        int slot = b * K_SEL + rank;
        out_b[slot] = (float)b;
        out_f[slot] = (float)n;
        out_a[slot] = (float)a;
        slot_src[slot] = b * N_LOC + n;
        int p = atomicAdd(&acnt[a], 1);
        aslot[a * TOTAL_SEL + p] = slot;
    }
}

/* ---- kernel 4: per-anchor grouped GEMM (fp32 WMMA) + bias, scatter to slots ---- */
__global__ __launch_bounds__(512) void k_post(const float* __restrict__ feats,
                                              const float* __restrict__ w_post,
                                              const float* __restrict__ b_post,
                                              const int* __restrict__ acnt,
                                              const int* __restrict__ aslot,
                                              const int* __restrict__ slot_src,
                                              float* __restrict__ out) {
    __shared__ float xt[16 * 260];   /* 260-float pitch: bank-skewed A fragments */
    __shared__ int s_slot[16];
    int a   = blockIdx.y;
    int cnt = acnt[a];
    int m0  = blockIdx.x * 16;
    if (m0 >= cnt) return;
    if (threadIdx.x < 16) {
        int mt = m0 + (int)threadIdx.x;
        s_slot[threadIdx.x] = (mt < cnt) ? aslot[a * TOTAL_SEL + mt] : -1;
    }
    __syncthreads();
    {   /* stage gathered X tile: 512 threads, 16 rows x 256 floats */
        int r = threadIdx.x >> 5;
        int l = threadIdx.x & 31;
        int slot = s_slot[r];
        if (slot >= 0) {
            const float* src = feats + (size_t)slot_src[slot] * C_DIM;
            *(float4*)&xt[r * 260 + l * 4]       = *(const float4*)(src + l * 4);
            *(float4*)&xt[r * 260 + 128 + l * 4] = *(const float4*)(src + 128 + l * 4);
        } else {
            float4 z = {0.f, 0.f, 0.f, 0.f};
            *(float4*)&xt[r * 260 + l * 4]       = z;
            *(float4*)&xt[r * 260 + 128 + l * 4] = z;
        }
    }
    __syncthreads();

    int wave  = threadIdx.x >> 5;   /* n-tile 0..15 */
    int lane  = threadIdx.x & 31;
    int mrow  = lane & 15;
    int khalf = lane >> 4;
    int ncol  = wave * 16 + mrow;
    const float* wbase = w_post + (size_t)a * C_DIM * C_DIM + ncol;

    v8f acc = {0.f, 0.f, 0.f, 0.f, 0.f, 0.f, 0.f, 0.f};
    for (int k0 = 0; k0 < C_DIM; k0 += 4) {
        int kb = k0 + 2 * khalf;
        v2f av = *(const v2f*)&xt[mrow * 260 + kb];
        v2f bv;
        bv.x = wbase[(size_t)kb * C_DIM];
        bv.y = wbase[(size_t)(kb + 1) * C_DIM];
        acc = __builtin_amdgcn_wmma_f32_16x16x4_f32(false, av, false, bv,
                                                    (short)0, acc, false, false);
    }
    float bias = b_post[a * C_DIM + ncol];
    for (int v = 0; v < 8; v++) {
        int slot = s_slot[v + 8 * khalf];
        if (slot >= 0) out[(size_t)slot * C_DIM + ncol] = acc[v] + bias;
    }
}

extern "C" void kernel_launch(void* const* d_in, const int* in_sizes, int n_in,
                              void* d_out, int out_size, void* d_ws, size_t ws_size,
                              hipStream_t stream) {
    (void)in_sizes; (void)n_in; (void)out_size; (void)ws_size;
    const float* feats    = (const float*)d_in[0];
    const float* w_logits = (const float*)d_in[1];
    const float* b_logits = (const float*)d_in[2];
    const float* w_post   = (const float*)d_in[3];
    const float* b_post   = (const float*)d_in[4];
    float* out = (float*)d_out;

    char* ws = (char*)d_ws;
    unsigned* keys   = (unsigned*)(ws + KEYS_OFF);
    unsigned* kth    = (unsigned*)(ws + KTH_OFF);
    int* acnt        = (int*)(ws + ACNT_OFF);
    int* slot_src    = (int*)(ws + SRC_OFF);
    int* aslot       = (int*)(ws + ASLOT_OFF);

    k_init<<<1, 32, 0, stream>>>(acnt);
    k_logits<<<NTILES / 8, 256, 0, stream>>>(feats, w_logits, b_logits, keys);
    k_select<<<B_DIM, 1024, 0, stream>>>(keys, kth);
    k_compact<<<B_DIM, 1024, 0, stream>>>(keys, kth, acnt, aslot, slot_src,
                                          out + OUT_B_OFF, out + OUT_F_OFF, out + OUT_A_OFF);
    k_post<<<dim3((TOTAL_SEL + 15) / 16, A_DIM), 512, 0, stream>>>(
        feats, w_post, b_post, acnt, aslot, slot_src, out);
}